// GAT_40913858461760
// MI455X (gfx1250) — compile-verified
//
#include <hip/hip_runtime.h>
#include <math.h>

// ---------------- problem constants (match reference) ----------------
#define NN   100000
#define EE   1600000
#define INF_ 128
#define HID  32
#define NH   4
#define NC   16
#define NEG_SLOPE 0.2f

typedef float v2f __attribute__((ext_vector_type(2)));
typedef float v8f __attribute__((ext_vector_type(8)));

// -------- order-preserving float <-> uint key (for atomic segment-max) ------
__device__ __forceinline__ unsigned fkey(float f) {
    unsigned u = __float_as_uint(f);
    return (u & 0x80000000u) ? ~u : (u | 0x80000000u);
}
__device__ __forceinline__ float fdecode(unsigned k) {
    unsigned u = (k & 0x80000000u) ? (k & 0x7FFFFFFFu) : ~k;
    return __uint_as_float(u);   // key 0 -> NaN (empty segment sentinel)
}

__device__ __forceinline__ float elu1(float x) { return x > 0.f ? x : expm1f(x); }
__device__ __forceinline__ float lrelu(float x) { return x > 0.f ? x : NEG_SLOPE * x; }

// ---------------- WMMA f32 GEMM: C[M,Nout] = A[M,128] @ B[128,Nout] ---------
// One block per 16-row tile. The 16x128 A panel is staged once in LDS with
// coalesced float4 loads; wave w computes the 16x16 tile at columns [16w,16w+16).
// V_WMMA_F32_16X16X4_F32 layouts (ISA 7.12.2):
//   A 16x4 : lane(0-15) M=lane, VGPRv K=v   ; lane(16-31) M=lane-16, K=2+v
//   B 4x16 : lane(0-15) N=lane, VGPRv K=v   ; lane(16-31) N=lane-16, K=2+v
//   C 16x16: VGPR r -> M=r (lanes 0-15) / M=8+r (lanes 16-31), N=lane&15
__global__ __launch_bounds__(256)
void k_gemm_wmma_f32(const float* __restrict__ A, const float* __restrict__ B,
                     float* __restrict__ C, int Nout) {
    __shared__ float As[16 * INF_];                 // 8 KB A panel
    const int tid  = threadIdx.x;
    const int row0 = blockIdx.x << 4;

    // ---- stage A panel to LDS (coalesced b128) ----
    for (int idx = tid * 4; idx < 16 * INF_; idx += blockDim.x * 4) {
        int r = idx >> 7, c = idx & (INF_ - 1);
        *(float4*)&As[idx] = *(const float4*)&A[(size_t)(row0 + r) * INF_ + c];
    }
    __syncthreads();

    const int wave = tid >> 5, lane = tid & 31;
    const int col0 = wave << 4;
    if (col0 < Nout) {
        const int half = lane >> 4;
        const int m    = lane & 15;
        v8f acc = {};
        const float* Ar = &As[m * INF_ + 2 * half];   // LDS (shared across waves)
        const float* Bp = B + col0 + m;               // global (L2-resident weights)
        for (int k0 = 0; k0 < INF_; k0 += 4) {
            v2f a, b;
            a.x = Ar[k0];
            a.y = Ar[k0 + 1];
            b.x = Bp[(size_t)(k0 + 2 * half)     * Nout];
            b.y = Bp[(size_t)(k0 + 2 * half + 1) * Nout];
            acc = __builtin_amdgcn_wmma_f32_16x16x4_f32(
                      false, a, false, b, (short)0, acc, false, false);
        }
        float* Cp = C + (size_t)row0 * Nout + col0 + m;
#pragma unroll
        for (int r = 0; r < 8; ++r)
            Cp[(size_t)(r + 8 * half) * Nout] = acc[r];
    }
}

// -------- per-node attention projections: el/er[n,h] = <feat[n,h,:], al/ar[h,:]>
__global__ void k_attn_proj(const float* __restrict__ feat,
                            const float* __restrict__ al,
                            const float* __restrict__ ar,
                            float* __restrict__ el, float* __restrict__ er,
                            int n, int heads, int outf) {
    int t = blockIdx.x * blockDim.x + threadIdx.x;
    if (t >= n * heads) return;
    int node = t / heads, h = t - node * heads;
    const float* f  = feat + (size_t)node * heads * outf + (size_t)h * outf;
    const float* ap = al + (size_t)h * outf;
    const float* bp = ar + (size_t)h * outf;
    float sl = 0.f, sr = 0.f;
    for (int i = 0; i < outf; i += 4) {
        float4 fv = *(const float4*)&f[i];
        float4 av = *(const float4*)&ap[i];
        float4 bv = *(const float4*)&bp[i];
        sl += fv.x * av.x + fv.y * av.y + fv.z * av.z + fv.w * av.w;
        sr += fv.x * bv.x + fv.y * bv.y + fv.z * bv.z + fv.w * bv.w;
    }
    el[t] = sl; er[t] = sr;
}

// -------- edge pass 1: e = lrelu(el[src]+er[dst]); stash e; atomic segment max
__global__ void k_edge_logit_max(const int* __restrict__ src,
                                 const int* __restrict__ dst,
                                 const float* __restrict__ el,
                                 const float* __restrict__ er,
                                 float* __restrict__ ebuf,
                                 unsigned* __restrict__ emaxu,
                                 int ne, int heads) {
    int i = blockIdx.x * blockDim.x + threadIdx.x;
    if (i >= ne) return;
    int s = src[i], d = dst[i];
    if (heads == 4) {
        float4 a = *(const float4*)&el[s * 4];
        float4 b = *(const float4*)&er[d * 4];
        float4 e;
        e.x = lrelu(a.x + b.x); e.y = lrelu(a.y + b.y);
        e.z = lrelu(a.z + b.z); e.w = lrelu(a.w + b.w);
        *(float4*)&ebuf[(size_t)i * 4] = e;
        atomicMax(&emaxu[d * 4 + 0], fkey(e.x));
        atomicMax(&emaxu[d * 4 + 1], fkey(e.y));
        atomicMax(&emaxu[d * 4 + 2], fkey(e.z));
        atomicMax(&emaxu[d * 4 + 3], fkey(e.w));
    } else {
        float e = lrelu(el[s] + er[d]);
        ebuf[i] = e;
        atomicMax(&emaxu[d], fkey(e));
    }
}

// -------- edge pass 2: w = exp(e - emax[dst]); stash w; atomic segment sum
__global__ void k_edge_exp_sum(const int* __restrict__ dst,
                               float* __restrict__ ebuf,
                               const unsigned* __restrict__ emaxu,
                               float* __restrict__ z,
                               int ne, int heads) {
    int i = blockIdx.x * blockDim.x + threadIdx.x;
    if (i >= ne) return;
    int d = dst[i];
    if (heads == 4) {
        float4 e = *(const float4*)&ebuf[(size_t)i * 4];
        uint4  k = *(const uint4*)&emaxu[d * 4];
        float m0 = fdecode(k.x); if (!isfinite(m0)) m0 = 0.f;
        float m1 = fdecode(k.y); if (!isfinite(m1)) m1 = 0.f;
        float m2 = fdecode(k.z); if (!isfinite(m2)) m2 = 0.f;
        float m3 = fdecode(k.w); if (!isfinite(m3)) m3 = 0.f;
        float4 w;
        w.x = expf(e.x - m0); w.y = expf(e.y - m1);
        w.z = expf(e.z - m2); w.w = expf(e.w - m3);
        *(float4*)&ebuf[(size_t)i * 4] = w;
        atomicAdd(&z[d * 4 + 0], w.x);
        atomicAdd(&z[d * 4 + 1], w.y);
        atomicAdd(&z[d * 4 + 2], w.z);
        atomicAdd(&z[d * 4 + 3], w.w);
    } else {
        float m = fdecode(emaxu[d]);
        if (!isfinite(m)) m = 0.f;
        float w = expf(ebuf[i] - m);
        ebuf[i] = w;
        atomicAdd(&z[d], w);
    }
}

// -------- edge pass 3 (wave per edge): a = w/z[dst] (final output),
//          rst[dst,:] += feat[src,:] * a[head]   (float4 gather, atomic scatter)
__global__ __launch_bounds__(256)
void k_edge_agg(const int* __restrict__ src, const int* __restrict__ dst,
                float* __restrict__ abuf, const float* __restrict__ z,
                const float* __restrict__ feat, float* __restrict__ rst,
                int ne, int heads, int outf) {
    int edge = (blockIdx.x * blockDim.x + threadIdx.x) >> 5;
    int lane = threadIdx.x & 31;
    if (edge >= ne) return;
    int s = src[edge], d = dst[edge];
    const int ft = heads * outf;
    const float* fs = feat + (size_t)s * ft;
    float*       rd = rst  + (size_t)d * ft;
    __builtin_prefetch(fs, 0, 0);                     // global_prefetch_b8

    float an[NH];
    for (int h = 0; h < heads; ++h)
        an[h] = abuf[(size_t)edge * heads + h] / z[d * heads + h];
    if (lane < heads) abuf[(size_t)edge * heads + lane] = an[lane];

    int f0 = lane * 4;                                // 4 floats share one head
    if (f0 < ft) {
        int h = f0 / outf;
        float4 v = *(const float4*)&fs[f0];
        atomicAdd(&rd[f0 + 0], v.x * an[h]);
        atomicAdd(&rd[f0 + 1], v.y * an[h]);
        atomicAdd(&rd[f0 + 2], v.z * an[h]);
        atomicAdd(&rd[f0 + 3], v.w * an[h]);
    }
}

// ---------------- elementwise helpers ----------------
__global__ void k_elu(const float* __restrict__ in, float* __restrict__ out, size_t n) {
    size_t i = (size_t)blockIdx.x * blockDim.x + threadIdx.x;
    if (i < n) out[i] = elu1(in[i]);
}
__global__ void k_res_elu2(const float* __restrict__ rst, const float* __restrict__ h,
                           float* __restrict__ out, size_t n) {
    size_t i = (size_t)blockIdx.x * blockDim.x + threadIdx.x;
    if (i < n) out[i] = elu1(elu1(rst[i] + h[i]));
}
__global__ void k_add(const float* __restrict__ a, const float* __restrict__ b,
                      float* __restrict__ out, size_t n) {
    size_t i = (size_t)blockIdx.x * blockDim.x + threadIdx.x;
    if (i < n) out[i] = a[i] + b[i];
}

// ---------------- host orchestration ----------------
static inline unsigned gdiv(size_t n, unsigned b) { return (unsigned)((n + b - 1) / b); }

extern "C" void kernel_launch(void* const* d_in, const int* in_sizes, int n_in,
                              void* d_out, int out_size, void* d_ws, size_t ws_size,
                              hipStream_t stream) {
    (void)in_sizes; (void)n_in; (void)out_size; (void)ws_size;
    const float* x     = (const float*)d_in[0];
    const int*   src   = (const int*)  d_in[1];
    const int*   dst   = (const int*)  d_in[2];
    const float* W0    = (const float*)d_in[3];
    const float* al0   = (const float*)d_in[4];
    const float* ar0   = (const float*)d_in[5];
    const float* W1    = (const float*)d_in[6];
    const float* al1   = (const float*)d_in[7];
    const float* ar1   = (const float*)d_in[8];
    const float* W2    = (const float*)d_in[9];
    const float* al2   = (const float*)d_in[10];
    const float* ar2   = (const float*)d_in[11];
    const float* Wres2 = (const float*)d_in[12];

    float* out    = (float*)d_out;
    float* logits = out;                               // [N,16]
    float* a0     = out + (size_t)NN * NC;             // [E,4]
    float* a1     = a0  + (size_t)EE * NH;             // [E,4]
    float* a2     = a1  + (size_t)EE * NH;             // [E,1]

    const size_t NF = (size_t)NN * (NH * HID);         // 12.8M floats
    float*    ws    = (float*)d_ws;
    float*    feat  = ws;                              // [N,128] / L2 smalls
    float*    hbuf  = ws + NF;                         // h1 then h2
    float*    rstb  = ws + 2 * NF;                     // aggregation target
    float*    el    = ws + 3 * NF;                     // [N,4]
    float*    er    = el + (size_t)NN * NH;
    float*    zb    = er + (size_t)NN * NH;
    unsigned* emaxu = (unsigned*)(zb + (size_t)NN * NH);
    float*    feat2 = feat;                            // [N,16]
    float*    res2  = feat + (size_t)NN * NC;          // [N,16]
    float*    rst2  = feat + 2 * (size_t)NN * NC;      // [N,16]

    const unsigned gRowT  = NN / 16;                   // 6250 row tiles
    const unsigned gNode4 = gdiv((size_t)NN * NH, 256);
    const unsigned gNode1 = gdiv((size_t)NN, 256);
    const unsigned gEdge  = gdiv((size_t)EE, 256);
    const unsigned gEdgeW = gdiv((size_t)EE * 32, 256);

    // ================= layer 0: x -> h1 =================
    k_gemm_wmma_f32<<<gRowT, 256, 0, stream>>>(x, W0, feat, NH * HID);
    k_attn_proj<<<gNode4, 256, 0, stream>>>(feat, al0, ar0, el, er, NN, NH, HID);
    hipMemsetAsync(emaxu, 0, (size_t)NN * NH * 4, stream);   // key 0 == -inf sentinel
    hipMemsetAsync(zb,    0, (size_t)NN * NH * 4, stream);
    hipMemsetAsync(rstb,  0, NF * 4, stream);
    k_edge_logit_max<<<gEdge, 256, 0, stream>>>(src, dst, el, er, a0, emaxu, EE, NH);
    k_edge_exp_sum <<<gEdge, 256, 0, stream>>>(dst, a0, emaxu, zb, EE, NH);
    k_edge_agg     <<<gEdgeW, 256, 0, stream>>>(src, dst, a0, zb, feat, rstb, EE, NH, HID);
    k_elu<<<gdiv(NF, 256), 256, 0, stream>>>(rstb, hbuf, NF);

    // ================= layer 1: h1 -> h2 (identity residual, elu(elu(.))) ====
    k_gemm_wmma_f32<<<gRowT, 256, 0, stream>>>(hbuf, W1, feat, NH * HID);
    k_attn_proj<<<gNode4, 256, 0, stream>>>(feat, al1, ar1, el, er, NN, NH, HID);
    hipMemsetAsync(emaxu, 0, (size_t)NN * NH * 4, stream);
    hipMemsetAsync(zb,    0, (size_t)NN * NH * 4, stream);
    hipMemsetAsync(rstb,  0, NF * 4, stream);
    k_edge_logit_max<<<gEdge, 256, 0, stream>>>(src, dst, el, er, a1, emaxu, EE, NH);
    k_edge_exp_sum <<<gEdge, 256, 0, stream>>>(dst, a1, emaxu, zb, EE, NH);
    k_edge_agg     <<<gEdgeW, 256, 0, stream>>>(src, dst, a1, zb, feat, rstb, EE, NH, HID);
    k_res_elu2<<<gdiv(NF, 256), 256, 0, stream>>>(rstb, hbuf, hbuf, NF);   // h2 in place

    // ================= layer 2: h2 -> logits (linear residual) =============
    k_gemm_wmma_f32<<<gRowT, 64, 0, stream>>>(hbuf, W2,    feat2, NC);
    k_gemm_wmma_f32<<<gRowT, 64, 0, stream>>>(hbuf, Wres2, res2,  NC);
    k_attn_proj<<<gNode1, 256, 0, stream>>>(feat2, al2, ar2, el, er, NN, 1, NC);
    hipMemsetAsync(emaxu, 0, (size_t)NN * 4, stream);
    hipMemsetAsync(zb,    0, (size_t)NN * 4, stream);
    hipMemsetAsync(rst2,  0, (size_t)NN * NC * 4, stream);
    k_edge_logit_max<<<gEdge, 256, 0, stream>>>(src, dst, el, er, a2, emaxu, EE, 1);
    k_edge_exp_sum <<<gEdge, 256, 0, stream>>>(dst, a2, emaxu, zb, EE, 1);
    k_edge_agg     <<<gEdgeW, 256, 0, stream>>>(src, dst, a2, zb, feat2, rst2, EE, 1, NC);
    k_add<<<gdiv((size_t)NN * NC, 256), 256, 0, stream>>>(rst2, res2, logits, (size_t)NN * NC);
}